// WDECTClassifier_27401891348672
// MI455X (gfx1250) — compile-verified
//
#include <hip/hip_runtime.h>
#include <hip/hip_bf16.h>

typedef float v2f __attribute__((ext_vector_type(2)));
typedef float v8f __attribute__((ext_vector_type(8)));

#define N_NODES 16000
#define N_EDGES 48000
#define N_FACES 32000
#define NDIR    64
#define BUMPS   32
#define NG      32
#define HID     256
#define NC      10
#define K1      (BUMPS * NDIR)   // 2048
#define SIG_SCALE 100.0f
#define CHUNK   512

// sigmoid(t) = 0.5 + 0.5*tanh(t/2); gfx1250 has a single-instruction v_tanh_f32.
__device__ __forceinline__ float signed_sigmoid(float t, float half_sign) {
#if __has_builtin(__builtin_amdgcn_tanhf)
    return fmaf(__builtin_amdgcn_tanhf(0.5f * t), half_sign, half_sign);
#elif __has_builtin(__builtin_amdgcn_tanh_f32)
    return fmaf(__builtin_amdgcn_tanh_f32(0.5f * t), half_sign, half_sign);
#else
    return (2.0f * half_sign) / (1.0f + __expf(-t));
#endif
}

// ---------------- zero accumulator ----------------
__global__ void ecc_zero_kernel(float* __restrict__ p, int n) {
    int i = blockIdx.x * blockDim.x + threadIdx.x;
    if (i < n) p[i] = 0.0f;
}

// ---------------- node heights: nh[i][d] = w[i] * (x[i] . v[:,d]) ----------------
__global__ __launch_bounds__(256) void nh_kernel(
    const float* __restrict__ x, const float* __restrict__ nw,
    const float* __restrict__ v, float* __restrict__ nh) {
    int i = blockIdx.x * blockDim.x + threadIdx.x;
    if (i >= N_NODES * NDIR) return;
    int node = i >> 6;
    int d    = i & 63;
    float w = nw[node];
    float h = w * (x[node * 3 + 0] * v[d] +
                   x[node * 3 + 1] * v[NDIR + d] +
                   x[node * 3 + 2] * v[2 * NDIR + d]);
    nh[i] = h;
}

// ---------------- ECC accumulation with 64KB LDS accumulator ----------------
// grid.x = element chunks over [nodes | edges | faces], grid.y = 4 bump chunks of 8
__global__ __launch_bounds__(256) void ecc_kernel(
    const float* __restrict__ nh, const float* __restrict__ ew,
    const float* __restrict__ fw, const int* __restrict__ ei,
    const int* __restrict__ fc, const int* __restrict__ batch,
    float* __restrict__ eccg) {
    __shared__ float acc[NG * 8 * NDIR];   // 32*8*64 floats = 64 KB in 320KB WGP LDS
    int tid = threadIdx.x;
    for (int i = tid; i < NG * 8 * NDIR; i += 256) acc[i] = 0.0f;
    __syncthreads();

    const int group  = tid >> 6;      // 4 element-lanes per block
    const int d      = tid & 63;      // direction
    const int base   = blockIdx.x * CHUNK;
    const int bchunk = blockIdx.y;    // bumps [bchunk*8, bchunk*8+8)
    const int M = N_NODES + N_EDGES + N_FACES;
    const int lim = (base + CHUNK < M) ? base + CHUNK : M;

    for (int mi = base + group; mi < lim; mi += 4) {
        // prefetch next element's gather rows (global_prefetch_b8)
        int mn = mi + 4;
        if (mn < lim) {
            if (mn < N_NODES) {
                __builtin_prefetch(&nh[mn * NDIR + d], 0, 0);
            } else if (mn < N_NODES + N_EDGES) {
                int e = mn - N_NODES;
                __builtin_prefetch(&nh[ei[e] * NDIR + d], 0, 0);
                __builtin_prefetch(&nh[ei[N_EDGES + e] * NDIR + d], 0, 0);
            } else {
                int f = mn - N_NODES - N_EDGES;
                __builtin_prefetch(&nh[fc[f] * NDIR + d], 0, 0);
                __builtin_prefetch(&nh[fc[N_FACES + f] * NDIR + d], 0, 0);
                __builtin_prefetch(&nh[fc[2 * N_FACES + f] * NDIR + d], 0, 0);
            }
        }

        float hv, half_sign;
        int g;
        if (mi < N_NODES) {
            hv = nh[mi * NDIR + d];
            g = batch[mi];
            half_sign = 0.5f;
        } else if (mi < N_NODES + N_EDGES) {
            int e = mi - N_NODES;
            int a = ei[e], b = ei[N_EDGES + e];
            hv = fmaxf(nh[a * NDIR + d], nh[b * NDIR + d]) * ew[e];
            g = batch[a];
            half_sign = -0.5f;
        } else {
            int f = mi - N_NODES - N_EDGES;
            int i0 = fc[f], i1 = fc[N_FACES + f], i2 = fc[2 * N_FACES + f];
            hv = fmaxf(fmaxf(nh[i0 * NDIR + d], nh[i1 * NDIR + d]),
                       nh[i2 * NDIR + d]) * fw[f];
            g = batch[i0];
            half_sign = 0.5f;
        }
        #pragma unroll
        for (int bb = 0; bb < 8; ++bb) {
            int b = bchunk * 8 + bb;
            float lin = -1.0f + (2.0f / 31.0f) * (float)b;
            float t = SIG_SCALE * (lin - hv);
            float s = signed_sigmoid(t, half_sign);          // v_tanh_f32 path
            __hip_atomic_fetch_add(&acc[g * 512 + bb * 64 + d], s,
                                   __ATOMIC_RELAXED, __HIP_MEMORY_SCOPE_WORKGROUP);
        }
    }
    __syncthreads();

    // flush LDS accumulator to global ecc [g][b][d]
    for (int i = tid; i < NG * 8 * NDIR; i += 256) {
        int g   = i >> 9;
        int rem = i & 511;
        int bb  = rem >> 6;
        int dd  = rem & 63;
        float vv = acc[i];
        if (vv != 0.0f)
            __hip_atomic_fetch_add(&eccg[g * K1 + (bchunk * 8 + bb) * 64 + dd], vv,
                                   __ATOMIC_RELAXED, __HIP_MEMORY_SCOPE_AGENT);
    }
}

// ---------------- MLP head: fp32 WMMA 16x16x4 GEMM + relu + tiny layer 2 ----------------
// One workgroup, 32 waves. Wave w: mt = w>>4 (2 row tiles of 16 graphs),
// nt = w&15 (16 col tiles of 16 hidden units). K = 2048 in steps of 4.
__global__ __launch_bounds__(1024) void mlp_kernel(
    const float* __restrict__ flat, const float* __restrict__ W1,
    const float* __restrict__ b1, const float* __restrict__ W2,
    const float* __restrict__ b2, float* __restrict__ out) {
    __shared__ float hbuf[NG * HID];   // 32KB hidden activations

    int tid  = threadIdx.x;
    int lane = tid & 31;
    int wid  = tid >> 5;       // 0..31
    int mt   = wid >> 4;       // 0..1
    int nt   = wid & 15;       // 0..15

    // A 16x4 f32 layout: lanes 0-15 hold M=0..15 K={0,1}; lanes 16-31 K={2,3}
    // B 4x16  f32 layout: lanes 0-15 N=0..15 K={0,1} per VGPR; lanes 16-31 K={2,3}
    int mrow = mt * 16 + (lane & 15);
    int ncol = nt * 16 + (lane & 15);
    int koff = (lane >> 4) * 2;
    const float* arow = flat + mrow * K1 + koff;    // flat[g][k]
    const float* brow = W1   + ncol * K1 + koff;    // B[k][n] = W1[n][k]

    v8f c = {};
    for (int k0 = 0; k0 < K1; k0 += 4) {
        v2f a, b;
        a.x = arow[k0];     a.y = arow[k0 + 1];
        b.x = brow[k0];     b.y = brow[k0 + 1];
        c = __builtin_amdgcn_wmma_f32_16x16x4_f32(
                false, a, false, b, (short)0, c, false, false);
    }

    // C/D layout: VGPR r -> M = mt*16 + (lane>=16 ? 8 : 0) + r, N = nt*16 + (lane&15)
    int mbase = mt * 16 + (lane >> 4) * 8;
    float bias = b1[ncol];
    #pragma unroll
    for (int r = 0; r < 8; ++r) {
        float val = c[r] + bias;
        hbuf[(mbase + r) * HID + ncol] = fmaxf(val, 0.0f);
    }
    __syncthreads();

    // layer 2: logits[32][10]
    if (tid < NG * NC) {
        int g = tid / NC, cls = tid % NC;
        float sum = b2[cls];
        for (int k = 0; k < HID; ++k)
            sum += hbuf[g * HID + k] * W2[cls * HID + k];
        out[g * NC + cls] = sum;
    }

    // second tuple output: flat itself
    for (int i = tid; i < NG * K1; i += 1024)
        out[NG * NC + i] = flat[i];
}

extern "C" void kernel_launch(void* const* d_in, const int* in_sizes, int n_in,
                              void* d_out, int out_size, void* d_ws, size_t ws_size,
                              hipStream_t stream) {
    const float* x   = (const float*)d_in[0];
    const float* nw  = (const float*)d_in[1];
    const float* ew  = (const float*)d_in[2];
    const float* fw  = (const float*)d_in[3];
    const float* v   = (const float*)d_in[4];
    const float* W1  = (const float*)d_in[5];
    const float* b1  = (const float*)d_in[6];
    const float* W2  = (const float*)d_in[7];
    const float* b2  = (const float*)d_in[8];
    const int*   ei  = (const int*)d_in[9];
    const int*   fc  = (const int*)d_in[10];
    const int*   bid = (const int*)d_in[11];

    float* nh   = (float*)d_ws;                 // 16000*64 floats = 4.0 MB
    float* eccg = nh + N_NODES * NDIR;          // 32*2048 floats  = 256 KB
    float* out  = (float*)d_out;

    ecc_zero_kernel<<<(NG * K1 + 255) / 256, 256, 0, stream>>>(eccg, NG * K1);
    nh_kernel<<<(N_NODES * NDIR + 255) / 256, 256, 0, stream>>>(x, nw, v, nh);

    const int M = N_NODES + N_EDGES + N_FACES;
    dim3 eg((M + CHUNK - 1) / CHUNK, 4);
    ecc_kernel<<<eg, 256, 0, stream>>>(nh, ew, fw, ei, fc, bid, eccg);

    mlp_kernel<<<1, 1024, 0, stream>>>(eccg, W1, b1, W2, b2, out);
}